// _JREModule_11708080849421
// MI455X (gfx1250) — compile-verified
//
#include <hip/hip_runtime.h>
#include <math.h>

// ---------------------------------------------------------------------------
// Non-local attention block for MI455X (gfx1250, wave32, WMMA).
// fp32 WMMA (16x16x4) for reference-level accuracy; two-pass online softmax
// over the query axis (recompute f) avoids a 268MB f buffer; all GEMMs are
// blocked so every operand is read from HBM/L2 once (or a few times) and
// shared through LDS / registers. Global loads are batched ahead of the MMA
// loops so s_wait_loadcnt drains incrementally instead of stalling per-load.
// ---------------------------------------------------------------------------

typedef __attribute__((ext_vector_type(2))) float v2f;
typedef __attribute__((ext_vector_type(8))) float v8f;

#define BB   16
#define CC   256
#define ICH  128
#define NN   4096   // 64*64
#define NSS  1024   // 32*32

__device__ __forceinline__ v8f wmma4(v2f a, v2f b, v8f c) {
  // D = A(16x4 f32) * B(4x16 f32) + C(16x16 f32)
  return __builtin_amdgcn_wmma_f32_16x16x4_f32(
      /*neg_a=*/false, a, /*neg_b=*/false, b,
      /*c_mod=*/(short)0, c, /*reuse_a=*/false, /*reuse_b=*/false);
}

// ---------------------------------------------------------------------------
// Kernel 1: all three 1x1 conv projections in one pass.
// Workgroup = 64-column block of x; 8 waves each own one 16-row m-tile and
// compute it for g/theta/phi (B-fragments from LDS shared by all three).
// x tile staged in LDS (stride 80). A fragments for the whole k-chunk are
// batch-preloaded (24 v2f) and overlap the staging stores.
// grid = (N/64, B), block = 256.
// ---------------------------------------------------------------------------
__global__ __launch_bounds__(256) void proj_kernel(
    const float* __restrict__ x,
    const float* __restrict__ gw, const float* __restrict__ gb,
    const float* __restrict__ tw, const float* __restrict__ tb,
    const float* __restrict__ pw, const float* __restrict__ pbi,
    float* __restrict__ gout, float* __restrict__ tout, float* __restrict__ pout)
{
  __shared__ float xbuf[32 * 80];          // [k_local][n_local + pad]
  const int tid  = threadIdx.x;
  const int wave = tid >> 5;
  const int lane = tid & 31;
  const int half = lane >> 4;
  const int l16  = lane & 15;
  const int batch = blockIdx.y;
  const int n0 = blockIdx.x * 64;
  const int m0 = wave * 16;

  const float* xb = x + (size_t)batch * CC * NN;
  const float* wgt[3]  = {gw, tw, pw};
  const float* bias[3] = {gb, tb, pbi};
  float*       outp[3] = {gout, tout, pout};

  v8f acc[3][4];
#pragma unroll
  for (int p = 0; p < 3; ++p)
#pragma unroll
    for (int j = 0; j < 4; ++j)
#pragma unroll
      for (int v = 0; v < 8; ++v) acc[p][j][v] = 0.0f;

  for (int k0 = 0; k0 < CC; k0 += 32) {
    __syncthreads();
    // stage x[k0..k0+32) x 64 cols -> LDS (512 float4, 2 per thread)
#pragma unroll
    for (int q = tid; q < 512; q += 256) {
      const int kr = q >> 4;
      const int c4 = (q & 15) * 4;
      const float4 v = *(const float4*)(xb + (size_t)(k0 + kr) * NN + n0 + c4);
      *(float4*)&xbuf[kr * 80 + c4] = v;
    }
    // batch-preload this chunk's A fragments (issues as one load clause,
    // overlapped with the LDS staging above)
    v2f afrag[3][8];
#pragma unroll
    for (int p = 0; p < 3; ++p)
#pragma unroll
      for (int ki = 0; ki < 8; ++ki)
        afrag[p][ki] =
            *(const v2f*)(wgt[p] + (size_t)(m0 + l16) * CC + k0 + ki * 4 + 2 * half);
    __syncthreads();
#pragma unroll
    for (int ki = 0; ki < 8; ++ki) {
      const int ks = ki * 4 + 2 * half;           // local k pair base
      v2f bfrag[4];
#pragma unroll
      for (int j = 0; j < 4; ++j) {
        bfrag[j].x = xbuf[ks * 80 + j * 16 + l16];
        bfrag[j].y = xbuf[(ks + 1) * 80 + j * 16 + l16];
      }
#pragma unroll
      for (int p = 0; p < 3; ++p)
#pragma unroll
        for (int j = 0; j < 4; ++j)
          acc[p][j] = wmma4(afrag[p][ki], bfrag[j], acc[p][j]);
    }
  }

#pragma unroll
  for (int p = 0; p < 3; ++p) {
    float* ob = outp[p] + (size_t)batch * ICH * NN;
#pragma unroll
    for (int j = 0; j < 4; ++j)
#pragma unroll
      for (int v = 0; v < 8; ++v) {
        const int row = m0 + 8 * half + v;        // D layout: VGPR v -> row
        const int col = n0 + j * 16 + l16;        // lane -> col
        ob[(size_t)row * NN + col] = acc[p][j][v] + bias[p][row];
      }
  }
}

// ---------------------------------------------------------------------------
// Kernel 2: 2x2 max pool. g stored (s,i) for the y GEMM, phi stored (i,s)
// for the f GEMM. grid = B*IC*NS/256.
// ---------------------------------------------------------------------------
__global__ __launch_bounds__(256) void pool_kernel(
    const float* __restrict__ gfull, const float* __restrict__ pfull,
    float* __restrict__ gpt, float* __restrict__ phip)
{
  const int idx = blockIdx.x * 256 + threadIdx.x;   // over B*ICH*NSS
  const int b   = idx >> 17;
  const int rem = idx & ((1 << 17) - 1);
  const int i   = rem >> 10;
  const int s   = rem & 1023;
  const int hs = s >> 5, wsx = s & 31;
  const int nbase = hs * 128 + wsx * 2;
  const size_t base = (size_t)b * ICH * NN + (size_t)i * NN + nbase;
  const float gm = fmaxf(fmaxf(gfull[base], gfull[base + 1]),
                         fmaxf(gfull[base + 64], gfull[base + 65]));
  const float pm = fmaxf(fmaxf(pfull[base], pfull[base + 1]),
                         fmaxf(pfull[base + 64], pfull[base + 65]));
  gpt[(size_t)b * NSS * ICH + (size_t)s * ICH + i] = gm;   // (s,i)
  phip[(size_t)b * ICH * NSS + (size_t)i * NSS + s] = pm;  // (i,s)
}

// ---------------------------------------------------------------------------
// Kernel 3: softmax stats over the QUERY axis (online max/sum per key col).
// Wave owns 16 key columns; phi B-fragments live in 32 v2f registers (phi
// read exactly once per batch). Theta is staged 32 rows at a time (stride 40
// -> half-wave bank shift of 16: conflict-free) and shared by all 8 waves;
// two n-subtiles are computed per stage to halve barrier count.
// grid = (NS/128, B), block = 256.
// ---------------------------------------------------------------------------
__global__ __launch_bounds__(256) void softmax_stats_kernel(
    const float* __restrict__ theta, const float* __restrict__ phip,
    float* __restrict__ cmax, float* __restrict__ csum)
{
  __shared__ float tbuf[128 * 40];          // [k][n_local(32) + pad]
  const int tid  = threadIdx.x;
  const int wave = tid >> 5;
  const int lane = tid & 31;
  const int half = lane >> 4;
  const int l16  = lane & 15;
  const int batch = blockIdx.y;
  const int s0 = blockIdx.x * 128 + wave * 16;
  const int scol = s0 + l16;

  const float* th = theta + (size_t)batch * ICH * NN;
  const float* ph = phip + (size_t)batch * ICH * NSS;

  // preload phi B-fragments for this wave's 16 columns (K = 128 -> 32 steps)
  v2f pB[32];
#pragma unroll
  for (int ki = 0; ki < 32; ++ki) {
    const int ks = ki * 4 + 2 * half;
    pB[ki].x = ph[(size_t)ks * NSS + scol];
    pB[ki].y = ph[(size_t)(ks + 1) * NSS + scol];
  }

  const int kk = tid >> 1;                  // staging: thread -> theta channel
  const int sh = (tid & 1) * 16;

  float run_m = -1e30f, run_z = 0.0f;
  for (int nt = 0; nt < NN; nt += 32) {
    __syncthreads();
    {   // stage theta tile [k=0..128)[n_local=0..32) (contiguous b128 stores)
      const float* src = th + (size_t)kk * NN + nt + sh;
      *(float4*)&tbuf[kk * 40 + sh]      = *(const float4*)(src);
      *(float4*)&tbuf[kk * 40 + sh + 4]  = *(const float4*)(src + 4);
      *(float4*)&tbuf[kk * 40 + sh + 8]  = *(const float4*)(src + 8);
      *(float4*)&tbuf[kk * 40 + sh + 12] = *(const float4*)(src + 12);
    }
    __syncthreads();

#pragma unroll
    for (int sub = 0; sub < 2; ++sub) {
      v8f f;
#pragma unroll
      for (int v = 0; v < 8; ++v) f[v] = 0.0f;
#pragma unroll
      for (int ki = 0; ki < 32; ++ki) {
        const int ks = ki * 4 + 2 * half;
        v2f a;
        a.x = tbuf[ks * 40 + sub * 16 + l16];
        a.y = tbuf[(ks + 1) * 40 + sub * 16 + l16];
        f = wmma4(a, pB[ki], f);
      }
      float tmax = f[0];
#pragma unroll
      for (int v = 1; v < 8; ++v) tmax = fmaxf(tmax, f[v]);
      tmax = fmaxf(tmax, __shfl_xor(tmax, 16, 32));
      const float nm = fmaxf(run_m, tmax);
      float se = 0.0f;
#pragma unroll
      for (int v = 0; v < 8; ++v) se += __expf(f[v] - nm);
      se += __shfl_xor(se, 16, 32);
      run_z = run_z * __expf(run_m - nm) + se;
      run_m = nm;
    }
  }
  if (half == 0) {
    cmax[batch * NSS + scol] = run_m;
    csum[batch * NSS + scol] = run_z;
  }
}

// ---------------------------------------------------------------------------
// Kernel 4: y = softmax(f) * g. Theta A-fragments for the wave's 16 rows are
// held in registers for the whole kernel (theta read once per batch). Phi/g
// tiles are staged in LDS per s-tile and shared by all waves. P round-trips
// stride-18 LDS (aligned b64, conflict-free) for the D->A layout transpose.
// grid = (N/128, B), block = 256.
// ---------------------------------------------------------------------------
__global__ __launch_bounds__(256) void attn_y_kernel(
    const float* __restrict__ theta, const float* __restrict__ phip,
    const float* __restrict__ gpt,
    const float* __restrict__ cmax, const float* __restrict__ csum,
    float* __restrict__ y)
{
  __shared__ float lphi[128 * 20];          // [k][s_local + pad]
  __shared__ float lg[16 * 136];            // [s_local][i + pad]
  __shared__ float pbuf[8][16 * 18];        // per-wave P tile [n][s + pad]
  const int tid  = threadIdx.x;
  const int wave = tid >> 5;
  const int lane = tid & 31;
  const int half = lane >> 4;
  const int l16  = lane & 15;
  const int batch = blockIdx.y;
  const int n0 = blockIdx.x * 128 + wave * 16;
  const int nrow = n0 + l16;

  const float* th = theta + (size_t)batch * ICH * NN;
  const float* ph = phip + (size_t)batch * ICH * NSS;
  const float* gp = gpt + (size_t)batch * NSS * ICH;

  // theta A-fragments for this wave's 16 rows, all K (32 v2f registers)
  v2f tA[32];
#pragma unroll
  for (int ki = 0; ki < 32; ++ki) {
    const int ks = ki * 4 + 2 * half;
    tA[ki].x = th[(size_t)ks * NN + nrow];
    tA[ki].y = th[(size_t)(ks + 1) * NN + nrow];
  }

  v8f yacc[8];
#pragma unroll
  for (int j = 0; j < 8; ++j)
#pragma unroll
    for (int v = 0; v < 8; ++v) yacc[j][v] = 0.0f;

  float* pb = pbuf[wave];
  const int pk = tid >> 1, psh = (tid & 1) * 8;   // lphi staging map
  const int gs = tid >> 4, gih = (tid & 15) * 8;  // lg staging map

  for (int st = 0; st < NSS; st += 16) {
    __syncthreads();
    {   // stage phi tile [k][s_local] and g tile [s_local][i]
      const float* psrc = ph + (size_t)pk * NSS + st + psh;
      *(float4*)&lphi[pk * 20 + psh]     = *(const float4*)(psrc);
      *(float4*)&lphi[pk * 20 + psh + 4] = *(const float4*)(psrc + 4);
      const float* gsrc = gp + (size_t)(st + gs) * ICH + gih;
      *(float4*)&lg[gs * 136 + gih]     = *(const float4*)(gsrc);
      *(float4*)&lg[gs * 136 + gih + 4] = *(const float4*)(gsrc + 4);
    }
    __syncthreads();

    // --- f tile (16n x 16s): A from registers, B from LDS ---
    v8f f;
#pragma unroll
    for (int v = 0; v < 8; ++v) f[v] = 0.0f;
#pragma unroll
    for (int ki = 0; ki < 32; ++ki) {
      const int ks = ki * 4 + 2 * half;
      v2f b;
      b.x = lphi[ks * 20 + l16];
      b.y = lphi[(ks + 1) * 20 + l16];
      f = wmma4(tA[ki], b, f);
    }

    // --- P = exp(f - m_s)/Z_s -> per-wave LDS (row n, col s) ---
    const float ms = cmax[batch * NSS + st + l16];
    const float rz = 1.0f / csum[batch * NSS + st + l16];
#pragma unroll
    for (int v = 0; v < 8; ++v)
      pb[(v + 8 * half) * 18 + l16] = __expf(f[v] - ms) * rz;
    __syncthreads();

    // --- y tile += P(16x16, K=s) x g(16s x 128i) ---
#pragma unroll
    for (int kq = 0; kq < 16; kq += 4) {
      const int ks = kq + 2 * half;
      const v2f a = *(const v2f*)&pb[l16 * 18 + ks];   // aligned b64
#pragma unroll
      for (int j = 0; j < 8; ++j) {
        const int icol = j * 16 + l16;
        v2f b;
        b.x = lg[ks * 136 + icol];
        b.y = lg[(ks + 1) * 136 + icol];
        yacc[j] = wmma4(a, b, yacc[j]);
      }
    }
  }

  float* yb = y + (size_t)batch * NN * ICH;   // layout (n, i)
#pragma unroll
  for (int j = 0; j < 8; ++j)
#pragma unroll
    for (int v = 0; v < 8; ++v) {
      const int row = n0 + v + 8 * half;
      const int col = j * 16 + l16;
      yb[(size_t)row * ICH + col] = yacc[j][v];
    }
}

// ---------------------------------------------------------------------------
// Kernel 5: wy(C,N) = W_w(C,IC) x y^T + W_b. Workgroup owns a 64-column
// block and ALL 256 output channels (2 m-tiles per wave); the y tile is
// staged once in LDS (stride 132 -> aligned ds_load_b64 B-fragments, each
// feeding two WMMAs). Weight A-fragments preloaded in batches of 8 k-steps.
// grid = (N/64, B), block = 256.
// ---------------------------------------------------------------------------
__global__ __launch_bounds__(256) void wconv_kernel(
    const float* __restrict__ y, const float* __restrict__ Ww,
    const float* __restrict__ Wb, float* __restrict__ wy)
{
  __shared__ float ybuf[64 * 132];          // [n_local][i + pad]
  const int tid  = threadIdx.x;
  const int wave = tid >> 5;
  const int lane = tid & 31;
  const int half = lane >> 4;
  const int l16  = lane & 15;
  const int batch = blockIdx.y;
  const int n0 = blockIdx.x * 64;
  const int m0 = wave * 16;
  const float* yb = y + (size_t)batch * NN * ICH;

  {   // stage y tile (64 n x 128 i), 8 float4 per thread
    const int sn = tid >> 2;
    const int iq = (tid & 3) * 32;
#pragma unroll
    for (int c = 0; c < 8; ++c) {
      const float4 v = *(const float4*)(yb + (size_t)(n0 + sn) * ICH + iq + c * 4);
      *(float4*)&ybuf[sn * 132 + iq + c * 4] = v;
    }
  }
  __syncthreads();

  v8f acc[2][4];
#pragma unroll
  for (int h = 0; h < 2; ++h)
#pragma unroll
    for (int j = 0; j < 4; ++j)
#pragma unroll
      for (int v = 0; v < 8; ++v) acc[h][j][v] = 0.0f;

#pragma unroll
  for (int kc = 0; kc < 32; kc += 8) {
    // batch-preload A fragments for 8 k-steps (one load clause)
    v2f a0[8], a1[8];
#pragma unroll
    for (int u = 0; u < 8; ++u) {
      const int ks = (kc + u) * 4 + 2 * half;
      a0[u] = *(const v2f*)(Ww + (size_t)(m0 + l16) * ICH + ks);
      a1[u] = *(const v2f*)(Ww + (size_t)(m0 + 128 + l16) * ICH + ks);
    }
#pragma unroll
    for (int u = 0; u < 8; ++u) {
      const int ks = (kc + u) * 4 + 2 * half;
#pragma unroll
      for (int j = 0; j < 4; ++j) {
        const v2f b = *(const v2f*)&ybuf[(j * 16 + l16) * 132 + ks];
        acc[0][j] = wmma4(a0[u], b, acc[0][j]);
        acc[1][j] = wmma4(a1[u], b, acc[1][j]);
      }
    }
  }

  float* ob = wy + (size_t)batch * CC * NN;
#pragma unroll
  for (int h = 0; h < 2; ++h)
#pragma unroll
    for (int j = 0; j < 4; ++j)
#pragma unroll
      for (int v = 0; v < 8; ++v) {
        const int row = m0 + h * 128 + 8 * half + v;
        const int col = n0 + j * 16 + l16;
        ob[(size_t)row * NN + col] = acc[h][j][v] + Wb[row];
      }
}

// ---------------------------------------------------------------------------
// Kernel 6: BN batch stats per channel -> fused scale/shift (deterministic
// tree reduction, replay-stable). grid = (C), block = 256.
// ---------------------------------------------------------------------------
__global__ __launch_bounds__(256) void bn_stats_kernel(
    const float* __restrict__ wy, const float* __restrict__ gamma,
    const float* __restrict__ beta,
    float* __restrict__ ch_scale, float* __restrict__ ch_shift)
{
  __shared__ float rs[256], rq[256];
  const int o = blockIdx.x, t = threadIdx.x;
  float s = 0.0f, q = 0.0f;
  for (int b = 0; b < BB; ++b) {
    const float* p = wy + (size_t)b * CC * NN + (size_t)o * NN;
    for (int n = t; n < NN; n += 256) { const float v = p[n]; s += v; q += v * v; }
  }
  rs[t] = s; rq[t] = q;
  __syncthreads();
  for (int off = 128; off > 0; off >>= 1) {
    if (t < off) { rs[t] += rs[t + off]; rq[t] += rq[t + off]; }
    __syncthreads();
  }
  if (t == 0) {
    const float inv = 1.0f / (float)(BB * NN);
    const float mean = rs[0] * inv;
    const float var  = fmaxf(rq[0] * inv - mean * mean, 0.0f);
    const float sc   = gamma[o] * rsqrtf(var + 1e-5f);
    ch_scale[o] = sc;
    ch_shift[o] = beta[o] - mean * sc;
  }
}

// ---------------------------------------------------------------------------
// Kernel 7: out = wy*scale + shift + x (fully coalesced)
// ---------------------------------------------------------------------------
__global__ __launch_bounds__(256) void bn_apply_kernel(
    const float* __restrict__ wy, const float* __restrict__ x,
    const float* __restrict__ ch_scale, const float* __restrict__ ch_shift,
    float* __restrict__ out)
{
  const size_t idx = (size_t)blockIdx.x * 256 + threadIdx.x;
  const int o = (int)((idx >> 12) & 255);   // (idx / 4096) % 256
  out[idx] = wy[idx] * ch_scale[o] + ch_shift[o] + x[idx];
}

// ---------------------------------------------------------------------------
// Workspace layout (floats). wy overlays the dead gfull+phifull region.
// Total = 37,782,016 floats (~144.1 MiB).
// ---------------------------------------------------------------------------
static constexpr size_t OFF_THETA = 0;                       // 8,388,608
static constexpr size_t OFF_GFULL = 8388608;                 // 8,388,608
static constexpr size_t OFF_PFULL = 16777216;                // 8,388,608
static constexpr size_t OFF_WY    = OFF_GFULL;               // 16,777,216 (overlay)
static constexpr size_t OFF_GPT   = 25165824;                // 2,097,152
static constexpr size_t OFF_PHIP  = 27262976;                // 2,097,152
static constexpr size_t OFF_CMAX  = 29360128;                // 16,384
static constexpr size_t OFF_CSUM  = 29376512;                // 16,384
static constexpr size_t OFF_Y     = 29392896;                // 8,388,608
static constexpr size_t OFF_SCL   = 37781504;                // 256
static constexpr size_t OFF_SHF   = 37781760;                // 256

extern "C" void kernel_launch(void* const* d_in, const int* in_sizes, int n_in,
                              void* d_out, int out_size, void* d_ws, size_t ws_size,
                              hipStream_t stream) {
  const float* x       = (const float*)d_in[0];
  const float* g_w     = (const float*)d_in[1];
  const float* g_b     = (const float*)d_in[2];
  const float* theta_w = (const float*)d_in[3];
  const float* theta_b = (const float*)d_in[4];
  const float* phi_w   = (const float*)d_in[5];
  const float* phi_b   = (const float*)d_in[6];
  const float* W_w     = (const float*)d_in[7];
  const float* W_b     = (const float*)d_in[8];
  const float* gamma   = (const float*)d_in[9];
  const float* beta    = (const float*)d_in[10];

  float* ws       = (float*)d_ws;
  float* theta_ws = ws + OFF_THETA;
  float* gfull    = ws + OFF_GFULL;
  float* pfull    = ws + OFF_PFULL;
  float* wy       = ws + OFF_WY;
  float* gpt      = ws + OFF_GPT;
  float* phip     = ws + OFF_PHIP;
  float* cmax     = ws + OFF_CMAX;
  float* csum     = ws + OFF_CSUM;
  float* yws      = ws + OFF_Y;
  float* chs      = ws + OFF_SCL;
  float* chf      = ws + OFF_SHF;

  proj_kernel<<<dim3(64, BB), 256, 0, stream>>>(
      x, g_w, g_b, theta_w, theta_b, phi_w, phi_b, gfull, theta_ws, pfull);
  pool_kernel<<<8192, 256, 0, stream>>>(gfull, pfull, gpt, phip);
  softmax_stats_kernel<<<dim3(8, BB), 256, 0, stream>>>(theta_ws, phip, cmax, csum);
  attn_y_kernel<<<dim3(32, BB), 256, 0, stream>>>(theta_ws, phip, gpt, cmax, csum, yws);
  wconv_kernel<<<dim3(64, BB), 256, 0, stream>>>(yws, W_w, W_b, wy);
  bn_stats_kernel<<<CC, 256, 0, stream>>>(wy, gamma, beta, chs, chf);
  bn_apply_kernel<<<65536, 256, 0, stream>>>(wy, x, chs, chf, (float*)d_out);
}